// DeformableStripConv_33612414058591
// MI455X (gfx1250) — compile-verified
//
#include <hip/hip_runtime.h>
#include <math.h>

typedef __attribute__((ext_vector_type(2))) float v2f;
typedef __attribute__((ext_vector_type(4))) float v4f;
typedef __attribute__((ext_vector_type(8))) float v8f;

// Problem constants
#define BATCH 8
#define CIN   64
#define COUT  64
#define HDIM  128
#define WDIM  128
#define HW    (HDIM * WDIM)          // 16384
#define NPIX  (BATCH * HW)           // 131072
#define KTAP  3
#define KDIM  (2 * KTAP * CIN)       // 384 contraction
#define NCHUNK (KDIM / 4)            // 96 WMMA K-chunks
#define WPACK_FLOATS (NCHUNK * 4 * 32 * 2)   // 24576 floats = 96KB

// ---------------------------------------------------------------------------
// Kernel 1: pack w_h / w_v into B-fragment layout for V_WMMA_F32_16X16X4_F32.
// B frag (4x16): lane<16  -> vgpr0 = Wcat[kk+0][n*16+lane], vgpr1 = Wcat[kk+1][..]
//                lane>=16 -> vgpr0 = Wcat[kk+2][n*16+lane-16], vgpr1 = Wcat[kk+3][..]
// Row r of Wcat: path*192 + tap*64 + c ; Wcat[r][o] = w_path[o][c][tap].
// Stored as float2 records: wpack[(q*4 + n)*32 + lane] = {v0, v1}.
// ---------------------------------------------------------------------------
__global__ void pack_weights_kernel(const float* __restrict__ w_h,
                                    const float* __restrict__ w_v,
                                    float* __restrict__ wpack) {
    int t = blockIdx.x * blockDim.x + threadIdx.x;
    if (t >= NCHUNK * 4 * 32) return;
    int lane = t & 31;
    int n    = (t >> 5) & 3;
    int q    = t >> 7;                 // 0..95
    int path = q / 48;
    int qq   = q % 48;
    int tap  = qq >> 4;
    int cg   = qq & 15;
    int c0   = (cg << 2) + ((lane >> 4) << 1);
    int o    = (n << 4) + (lane & 15);
    const float* src = (path == 0) ? w_h : w_v;   // both (64,64,3) effective layout
    float v0 = src[(o * CIN + c0)     * KTAP + tap];
    float v1 = src[(o * CIN + c0 + 1) * KTAP + tap];
    wpack[2 * t]     = v0;
    wpack[2 * t + 1] = v1;
}

// ---------------------------------------------------------------------------
// Kernel 2: offset conv 3x3, Cin=64 -> 6 channels, pad=1, plus bias.
// One thread per pixel; weight fetches are wave-uniform -> scalar loads.
// offs layout: offs[p*6 + j], j=0..2 h_off taps, j=3..5 v_off taps.
// ---------------------------------------------------------------------------
__global__ void offset_conv_kernel(const float* __restrict__ x,
                                   const float* __restrict__ ow,
                                   const float* __restrict__ ob,
                                   float* __restrict__ offs) {
    int p = blockIdx.x * blockDim.x + threadIdx.x;
    if (p >= NPIX) return;
    int b = p >> 14;
    int r = p & (HW - 1);
    int h = r >> 7;
    int w = r & (WDIM - 1);

    float acc[6];
#pragma unroll
    for (int j = 0; j < 6; ++j) acc[j] = ob[j];

    const float* xb = x + ((size_t)b << 20);       // b * 64 * 16384
    for (int c = 0; c < CIN; ++c) {
        const float* pl = xb + (c << 14);
#pragma unroll
        for (int dy = -1; dy <= 1; ++dy) {
            int yy = h + dy;
            if ((unsigned)yy > (unsigned)(HDIM - 1)) continue;
#pragma unroll
            for (int dx = -1; dx <= 1; ++dx) {
                int xx = w + dx;
                if ((unsigned)xx > (unsigned)(WDIM - 1)) continue;
                float xv = pl[(yy << 7) + xx];
#pragma unroll
                for (int j = 0; j < 6; ++j)
                    acc[j] += xv * ow[((j * CIN + c) * 3 + (dy + 1)) * 3 + (dx + 1)];
            }
        }
    }
#pragma unroll
    for (int j = 0; j < 6; ++j) offs[p * 6 + j] = acc[j];
}

// ---------------------------------------------------------------------------
// Kernel 3: fused deformable gather + WMMA GEMM.
// One wave per tile of 16 consecutive-w pixels. Accumulates 16x64 outputs in
// 4 WMMA accumulators over K=384 (h-path ++ v-path), 4 K-values per WMMA.
// Each bilinear gather degenerates to a 2-point lerp (one integer coord).
// ---------------------------------------------------------------------------
__global__ __launch_bounds__(128)
void deform_wmma_kernel(const float* __restrict__ x,
                        const v2f*  __restrict__ wpack,
                        const float* __restrict__ offs,
                        float* __restrict__ out) {
    const int lane = threadIdx.x & 31;
    const int wave = blockIdx.x * (blockDim.x >> 5) + (threadIdx.x >> 5);
    const int tile = wave;                    // 8192 tiles
    const int p0   = tile << 4;
    const int b     = p0 >> 14;
    const int h     = (p0 >> 7) & (HDIM - 1);
    const int wbase = p0 & (WDIM - 1);
    const int col   = lane & 15;
    const int half  = lane >> 4;
    const int w     = wbase + col;
    const int p     = p0 + col;

    // Per-pixel offsets (each half-wave computes the same pixel's taps).
    float offv[6];
#pragma unroll
    for (int j = 0; j < 6; ++j) offv[j] = offs[p * 6 + j];

    // Tap descriptors: 2 sample addresses (element offset in channel plane)
    // and 2 lerp weights, with boundary validity folded into the weights.
    int   ad0[6], ad1[6];
    float wt0[6], wt1[6];
#pragma unroll
    for (int k = 0; k < KTAP; ++k) {
        // horizontal tap: px integer, py = h + h_off  (j = k)
        {
            float py  = (float)h + offv[k];
            int   px  = w + k - 1;
            float vx  = (px >= 0 && px <= WDIM - 1) ? 1.f : 0.f;
            int   xi  = px < 0 ? 0 : (px > WDIM - 1 ? WDIM - 1 : px);
            float y0f = floorf(py);
            float fy  = py - y0f;
            int   y0  = (int)y0f;
            float vy0 = (y0f >= 0.f  && y0f <= 127.f) ? 1.f : 0.f;
            float vy1 = (y0f >= -1.f && y0f <= 126.f) ? 1.f : 0.f;
            int   y0c = y0 < 0 ? 0 : (y0 > 127 ? 127 : y0);
            int   y1  = y0 + 1;
            int   y1c = y1 < 0 ? 0 : (y1 > 127 ? 127 : y1);
            wt0[k] = (1.f - fy) * vy0 * vx;
            wt1[k] = fy * vy1 * vx;
            ad0[k] = (y0c << 7) + xi;
            ad1[k] = (y1c << 7) + xi;
        }
        // vertical tap: py integer, px = w + v_off  (j = 3 + k)
        {
            int   py  = h + k - 1;
            float vy  = (py >= 0 && py <= HDIM - 1) ? 1.f : 0.f;
            int   yi  = py < 0 ? 0 : (py > HDIM - 1 ? HDIM - 1 : py);
            float pxf = (float)w + offv[3 + k];
            float x0f = floorf(pxf);
            float fx  = pxf - x0f;
            int   x0  = (int)x0f;
            float vx0 = (x0f >= 0.f  && x0f <= 127.f) ? 1.f : 0.f;
            float vx1 = (x0f >= -1.f && x0f <= 126.f) ? 1.f : 0.f;
            int   x0c = x0 < 0 ? 0 : (x0 > 127 ? 127 : x0);
            int   x1  = x0 + 1;
            int   x1c = x1 < 0 ? 0 : (x1 > 127 ? 127 : x1);
            wt0[3 + k] = (1.f - fx) * vx0 * vy;
            wt1[3 + k] = fx * vx1 * vy;
            ad0[3 + k] = (yi << 7) + x0c;
            ad1[3 + k] = (yi << 7) + x1c;
        }
    }

    v8f z = {0.f, 0.f, 0.f, 0.f, 0.f, 0.f, 0.f, 0.f};
    v8f acc0 = z, acc1 = z, acc2 = z, acc3 = z;

    const float* xb = x + ((size_t)b << 20);
    const int choff = half << 1;              // 0 or 2: this lane's channel pair

#pragma unroll
    for (int path = 0; path < 2; ++path) {
#pragma unroll
        for (int tap = 0; tap < KTAP; ++tap) {
            const int   j  = path * 3 + tap;
            const int   A0 = ad0[j], A1 = ad1[j];
            const float W0 = wt0[j], W1 = wt1[j];
            const int   qbase = path * 48 + tap * 16;
#pragma unroll 4
            for (int cg = 0; cg < 16; ++cg) {
                const int    c0  = (cg << 2) + choff;
                const float* pl  = xb + ((size_t)c0 << 14);
                const float* pl2 = pl + HW;
                // A fragment: 16x4, lane holds {A[M=col][kk+2*half], [kk+2*half+1]}
                float s0 = W0 * pl[A0]  + W1 * pl[A1];
                float s1 = W0 * pl2[A0] + W1 * pl2[A1];
                v2f  A  = {s0, s1};
                const v2f* wp = wpack + (((qbase + cg) << 7) >> 1) * 2 + lane; // q*128 + lane
                acc0 = __builtin_amdgcn_wmma_f32_16x16x4_f32(
                        false, A, false, wp[0],  (short)0, acc0, false, false);
                acc1 = __builtin_amdgcn_wmma_f32_16x16x4_f32(
                        false, A, false, wp[32], (short)0, acc1, false, false);
                acc2 = __builtin_amdgcn_wmma_f32_16x16x4_f32(
                        false, A, false, wp[64], (short)0, acc2, false, false);
                acc3 = __builtin_amdgcn_wmma_f32_16x16x4_f32(
                        false, A, false, wp[96], (short)0, acc3, false, false);
            }
        }
    }

    // Store: lane writes 8 consecutive w positions for output o = n*16 + col.
    // D layout: vgpr v of lanes<16 = pixel v; lanes>=16 = pixel v+8.
    const size_t outb = (((size_t)b << 6) << 14) + (size_t)((h << 7) + wbase + (half << 3));
    {
        v8f a = acc0;
        float* dst = out + outb + ((size_t)(0 * 16 + col) << 14);
        *(v4f*)dst       = __builtin_shufflevector(a, a, 0, 1, 2, 3);
        *(v4f*)(dst + 4) = __builtin_shufflevector(a, a, 4, 5, 6, 7);
    }
    {
        v8f a = acc1;
        float* dst = out + outb + ((size_t)(1 * 16 + col) << 14);
        *(v4f*)dst       = __builtin_shufflevector(a, a, 0, 1, 2, 3);
        *(v4f*)(dst + 4) = __builtin_shufflevector(a, a, 4, 5, 6, 7);
    }
    {
        v8f a = acc2;
        float* dst = out + outb + ((size_t)(2 * 16 + col) << 14);
        *(v4f*)dst       = __builtin_shufflevector(a, a, 0, 1, 2, 3);
        *(v4f*)(dst + 4) = __builtin_shufflevector(a, a, 4, 5, 6, 7);
    }
    {
        v8f a = acc3;
        float* dst = out + outb + ((size_t)(3 * 16 + col) << 14);
        *(v4f*)dst       = __builtin_shufflevector(a, a, 0, 1, 2, 3);
        *(v4f*)(dst + 4) = __builtin_shufflevector(a, a, 4, 5, 6, 7);
    }
}

// ---------------------------------------------------------------------------
extern "C" void kernel_launch(void* const* d_in, const int* in_sizes, int n_in,
                              void* d_out, int out_size, void* d_ws, size_t ws_size,
                              hipStream_t stream) {
    const float* x   = (const float*)d_in[0];
    const float* ow  = (const float*)d_in[1];
    const float* ob  = (const float*)d_in[2];
    const float* w_h = (const float*)d_in[3];
    const float* w_v = (const float*)d_in[4];
    float* out = (float*)d_out;

    float* wpack = (float*)d_ws;                                   // 96 KB
    float* offsb = (float*)((char*)d_ws + WPACK_FLOATS * sizeof(float)); // 3 MB

    // 1) Pack weights into WMMA B-fragment layout.
    {
        int n = NCHUNK * 4 * 32;
        pack_weights_kernel<<<(n + 255) / 256, 256, 0, stream>>>(w_h, w_v, wpack);
    }
    // 2) Offset conv.
    offset_conv_kernel<<<NPIX / 256, 256, 0, stream>>>(x, ow, ob, offsb);
    // 3) Fused gather + WMMA GEMM: 8192 tiles, 1 wave each, 4 waves/block.
    deform_wmma_kernel<<<(NPIX / 16) / 4, 128, 0, stream>>>(
        x, (const v2f*)wpack, offsb, out);
}